// IRGS_Trans_16363825398166
// MI455X (gfx1250) — compile-verified
//
#include <hip/hip_runtime.h>
#include <stdint.h>

// Problem constants (from reference setup_inputs)
#define B_    4
#define C_    128
#define H_    512
#define W_    512
#define P_    (H_*W_)        // 262144 pixels per image
#define NT    800            // n_tokens (segments per image)
#define ML    1024           // max_length
#define NC    5              // label classes
#define NS    (B_*NT)        // 3200 global segments

// Tiling for the feature segment-sum kernel
#define TPB    256           // threads per block (8 wave32)
#define CH     32            // channels privatized per workgroup -> 800*32*4 = 102.4 KB LDS
#define TP     8192          // pixels per tile
#define SEGCH  4096          // pixels per async-staged segment chunk (16 KB)
#define NCHUNK (TP/SEGCH)    // 2

#if defined(__HIP_DEVICE_COMPILE__) && \
    __has_builtin(__builtin_amdgcn_global_load_async_to_lds_b128) && \
    __has_builtin(__builtin_amdgcn_s_wait_asynccnt)
#define USE_ASYNC 1
#else
#define USE_ASYNC 0
#endif

#if USE_ASYNC
// Builtin prototype (from hipcc diagnostic): 
//   void __builtin_amdgcn_global_load_async_to_lds_b128(
//        int __vector(4) addrspace(1)* gsrc, int __vector(4) addrspace(3)* lds,
//        int imm_offset, int imm_cpol)
typedef int v4i_ __attribute__((vector_size(16)));
typedef __attribute__((address_space(1))) v4i_* as1_v4i;
typedef __attribute__((address_space(3))) v4i_* as3_v4i;

__device__ __forceinline__ void async_copy_b128(const void* gsrc, void* ldst) {
  __builtin_amdgcn_global_load_async_to_lds_b128(
      (as1_v4i)gsrc, (as3_v4i)ldst, 0, 0);
}
#endif

__global__ void k_zero(float* __restrict__ p, int n) {
  int i = blockIdx.x * blockDim.x + threadIdx.x;
  int stride = gridDim.x * blockDim.x;
  for (; i < n; i += stride) p[i] = 0.0f;
}

// Per-pixel pass over gts+segments: counts and class histogram.
// 2M u32 atomics into a 77 KB region -> fully L2-resident on 192MB L2.
__global__ void k_hist(const int* __restrict__ gts, const int* __restrict__ seg,
                       unsigned* __restrict__ cnt, unsigned* __restrict__ hist) {
  int i = blockIdx.x * blockDim.x + threadIdx.x;  // over B_*P_
  if (i >= B_ * P_) return;
  int b = i / P_;
  int s = seg[i] + b * NT;
  atomicAdd(&cnt[s], 1u);
  atomicAdd(&hist[s * NC + gts[i]], 1u);
}

// LDS-privatized segment sum of features.
// grid = (P_/TP, C_/CH, B_); each WG owns a full 800-segment x 32-channel
// accumulator in LDS (320KB/WGP makes this possible), streams its channel
// planes with coalesced float4 loads, and scatters via ds_add_f32.
// Segment indices are double-buffered into LDS with async global->LDS copies.
__global__ __launch_bounds__(TPB) void k_featsum(const float* __restrict__ feat,
                                                 const int* __restrict__ seg,
                                                 float* __restrict__ sums) {
  __shared__ float acc[CH * NT];          // 102400 B, layout [cc][s]: bank = s%64 (random)
  __shared__ int   segbuf[2][SEGCH];      // 2 x 16 KB double buffer

  const int tile = blockIdx.x;
  const int cg   = blockIdx.y;
  const int b    = blockIdx.z;
  const int tid  = threadIdx.x;
  const int c0   = cg * CH;
  const int p0   = tile * TP;

  for (int i = tid; i < CH * NT; i += TPB) acc[i] = 0.0f;

  const int* segp = seg + (size_t)b * P_ + p0;

  // Stage chunk 0 of segment indices into LDS.
#if USE_ASYNC
  #pragma unroll
  for (int k = 0; k < SEGCH / (TPB * 4); ++k) {        // 4 x b128 per thread
    int off = (k * TPB + tid) * 4;                     // int index (16B granules)
    async_copy_b128(segp + off, &segbuf[0][off]);
  }
#else
  for (int off = tid * 4; off < SEGCH; off += TPB * 4)
    *(int4*)&segbuf[0][off] = *(const int4*)&segp[off];
#endif

  for (int ch = 0; ch < NCHUNK; ++ch) {
    // Wait for this wave's staged data, then barrier: all waves' chunk `ch`
    // is in LDS, and everyone is done reading the buffer we stage into next.
#if USE_ASYNC
    __builtin_amdgcn_s_wait_asynccnt(0);
#endif
    __syncthreads();

    // Kick off chunk ch+1 while we process chunk ch.
    if (ch + 1 < NCHUNK) {
      const int* nsrc = segp + (ch + 1) * SEGCH;
      int nb = (ch + 1) & 1;
#if USE_ASYNC
      #pragma unroll
      for (int k = 0; k < SEGCH / (TPB * 4); ++k) {
        int off = (k * TPB + tid) * 4;
        async_copy_b128(nsrc + off, &segbuf[nb][off]);
      }
#else
      for (int off = tid * 4; off < SEGCH; off += TPB * 4)
        *(int4*)&segbuf[nb][off] = *(const int4*)&nsrc[off];
#endif
    }

    // Process chunk ch: each thread owns 4 consecutive pixels per sweep;
    // for each channel the 256 lanes read one contiguous 4KB span (coalesced).
    const int cb = ch & 1;
    const int pbase = p0 + ch * SEGCH;
    for (int pp = tid * 4; pp < SEGCH; pp += TPB * 4) {
      int4 s4 = *(const int4*)&segbuf[cb][pp];
      const float* fp = feat + (size_t)(b * C_ + c0) * P_ + pbase + pp;
      #pragma unroll 4
      for (int cc = 0; cc < CH; ++cc) {
        float4 f = *(const float4*)(fp + (size_t)cc * P_);
        unsafeAtomicAdd(&acc[cc * NT + s4.x], f.x);
        unsafeAtomicAdd(&acc[cc * NT + s4.y], f.y);
        unsafeAtomicAdd(&acc[cc * NT + s4.z], f.z);
        unsafeAtomicAdd(&acc[cc * NT + s4.w], f.w);
      }
    }
  }

  __syncthreads();

  // Flush private accumulator: 25600 f32 atomics/WG into the 1.6MB global
  // accumulator (stays resident in L2).
  for (int i = tid; i < CH * NT; i += TPB) {
    int cc = i / NT;
    int s  = i - cc * NT;
    unsafeAtomicAdd(&sums[(size_t)(b * NT + s) * C_ + c0 + cc], acc[i]);
  }
}

// tokens = sums / max(cnt,1); labels = argmax(hist) (first/smallest wins ties);
// pads = max_length - n_tokens. Padding regions were zeroed by k_zero.
__global__ void k_finalize(const float* __restrict__ sums,
                           const unsigned* __restrict__ cnt,
                           const unsigned* __restrict__ hist,
                           float* __restrict__ tokens,
                           float* __restrict__ labels,
                           float* __restrict__ pads) {
  int i = blockIdx.x * blockDim.x + threadIdx.x;
  if (i < NS * C_) {
    int sgl = i / C_;
    int c   = i - sgl * C_;
    int b   = sgl / NT;
    int t   = sgl - b * NT;
    float n = (float)cnt[sgl];
    tokens[((size_t)b * ML + t) * C_ + c] = sums[i] / fmaxf(n, 1.0f);
  }
  if (i < NS) {
    int b = i / NT;
    int t = i - b * NT;
    unsigned best = hist[i * NC];
    int arg = 0;
    #pragma unroll
    for (int k = 1; k < NC; ++k) {
      unsigned h = hist[i * NC + k];
      if (h > best) { best = h; arg = k; }   // strict > keeps smallest class on ties
    }
    labels[b * ML + t] = (float)arg;
  }
  if (i < B_) pads[i] = (float)(ML - NT);
}

extern "C" void kernel_launch(void* const* d_in, const int* in_sizes, int n_in,
                              void* d_out, int out_size, void* d_ws, size_t ws_size,
                              hipStream_t stream) {
  const float* feat = (const float*)d_in[0];
  const int*   gts  = (const int*)d_in[1];
  const int*   seg  = (const int*)d_in[2];

  // Workspace layout: sums[NS*C_] f32 | cnt[NS] u32 | hist[NS*NC] u32
  float*    sums = (float*)d_ws;
  unsigned* cnt  = (unsigned*)((char*)d_ws + (size_t)NS * C_ * 4);
  unsigned* hist = (unsigned*)((char*)d_ws + (size_t)NS * C_ * 4 + (size_t)NS * 4);

  float* out    = (float*)d_out;
  float* tokens = out;                          // B*ML*C_
  float* labels = out + (size_t)B_ * ML * C_;   // B*ML
  float* pads   = labels + (size_t)B_ * ML;     // B

  const int wsN = NS * C_ + NS + NS * NC;       // 428800 dwords; zero bits == 0.0f/0u
  k_zero<<<(wsN + 255) / 256, 256, 0, stream>>>((float*)d_ws, wsN);
  k_zero<<<(out_size + 255) / 256, 256, 0, stream>>>(out, out_size);

  k_hist<<<(B_ * P_ + 255) / 256, 256, 0, stream>>>(gts, seg, cnt, hist);

  dim3 grid(P_ / TP, C_ / CH, B_);              // 32 x 4 x 4 = 512 workgroups
  k_featsum<<<grid, TPB, 0, stream>>>(feat, seg, sums);

  k_finalize<<<(NS * C_ + 255) / 256, 256, 0, stream>>>(sums, cnt, hist,
                                                        tokens, labels, pads);
}